// RelationalGATLayer_55344948576381
// MI455X (gfx1250) — compile-verified
//
#include <hip/hip_runtime.h>
#include <hip/hip_bf16.h>

// ---------------------------------------------------------------------------
// Relational GAT layer for MI455X (gfx1250, wave32).
// h_r = x @ W_r done with V_WMMA_F32_16X16X4_F32 (full f32 precision: GEMM is
// ~5 GFLOP, totally compute-trivial; the workload is bound by the 1.6M-edge
// gather/scatter phase which lives in the 192MB L2).
// ---------------------------------------------------------------------------

#define NNODES 50000
#define NEDGES 1600000
#define INDIM  128
#define HEADS  4
#define HDIM   32
#define NREL   3
#define ODIM   128   // HEADS*HDIM

typedef __attribute__((ext_vector_type(2))) float v2f;
typedef __attribute__((ext_vector_type(8))) float v8f;

__device__ __forceinline__ float leaky(float v) {
  return v > 0.0f ? v : 0.2f * v;
}

// monotone float -> uint mapping so unsigned atomicMax == float max
__device__ __forceinline__ unsigned ford(float f) {
  unsigned u = __float_as_uint(f);
  return (u & 0x80000000u) ? ~u : (u | 0x80000000u);
}
__device__ __forceinline__ float funord(unsigned o) {
  unsigned u = (o & 0x80000000u) ? (o & 0x7fffffffu) : ~o;
  return __uint_as_float(u);
}

__device__ __forceinline__ void relw(const float* __restrict__ rw, float* w) {
  float w0 = rw[0], w1 = rw[1], w2 = rw[2];
  float m = fmaxf(w0, fmaxf(w1, w2));
  float e0 = __expf(w0 - m), e1 = __expf(w1 - m), e2 = __expf(w2 - m);
  float inv = 1.0f / (e0 + e1 + e2);
  w[0] = e0 * inv; w[1] = e1 * inv; w[2] = e2 * inv;
}

// ---------------------------------------------------------------------------
// 1) GEMM: hbuf[r][n][:] = x[n][:] @ W[r]
//    grid = (3125 row-tiles of 16, 3 relations), block = 256 (8 waves).
//    A-tile (16x128 f32, 8KB) staged in LDS, shared by all 8 waves; each wave
//    owns one 16-wide N-tile and runs 32 chained V_WMMA_F32_16X16X4_F32.
// ---------------------------------------------------------------------------
__global__ __launch_bounds__(256) void rgat_gemm(const float* __restrict__ x,
                                                 const float* __restrict__ W,
                                                 float* __restrict__ hbuf) {
  __shared__ float As[16 * INDIM];
  const int r    = blockIdx.y;
  const int row0 = blockIdx.x * 16;

  for (int i = threadIdx.x; i < 16 * INDIM; i += 256)
    As[i] = x[(size_t)row0 * INDIM + i];
  __syncthreads();

  const int lane  = threadIdx.x & 31;
  const int wavei = threadIdx.x >> 5;          // N-tile 0..7
  const int m     = lane & 15;                 // row (A) / col (B) within tile
  const int koff  = (lane >> 4) << 1;          // lanes 16-31 take K+2,K+3

  const float* __restrict__ B = W + (size_t)r * INDIM * ODIM + wavei * 16;

  v8f acc = {0.f, 0.f, 0.f, 0.f, 0.f, 0.f, 0.f, 0.f};
  #pragma unroll
  for (int k0 = 0; k0 < INDIM; k0 += 4) {
    const int k = k0 + koff;
    v2f a, b;
    a.x = As[m * INDIM + k];
    a.y = As[m * INDIM + k + 1];
    b.x = B[(size_t)k * ODIM + m];
    b.y = B[(size_t)(k + 1) * ODIM + m];
    acc = __builtin_amdgcn_wmma_f32_16x16x4_f32(
        /*neg_a=*/false, a, /*neg_b=*/false, b,
        /*c_mod=*/(short)0, acc, /*reuse_a=*/false, /*reuse_b=*/false);
  }

  // C/D layout: VGPR v: lanes 0-15 -> M=v, lanes 16-31 -> M=v+8; N = lane&15
  const int mbase = (lane >> 4) << 3;
  float* __restrict__ op =
      hbuf + ((size_t)r * NNODES + row0) * ODIM + wavei * 16 + m;
  #pragma unroll
  for (int v = 0; v < 8; v++)
    op[(size_t)(mbase + v) * ODIM] = acc[v];
}

// ---------------------------------------------------------------------------
// 2) attention logits: a_s/a_d[(r*N+n)*H+h] = <h_r[n, h*32:h*32+32], att[r][h]>
// ---------------------------------------------------------------------------
__global__ void rgat_att(const float* __restrict__ hbuf,
                         const float* __restrict__ att_src,
                         const float* __restrict__ att_dst,
                         float* __restrict__ a_s, float* __restrict__ a_d) {
  const int idx = blockIdx.x * blockDim.x + threadIdx.x;
  if (idx >= NREL * NNODES * HEADS) return;
  const int h  = idx & (HEADS - 1);
  const int rn = idx >> 2;                 // r*N + n
  const int r  = rn / NNODES;
  const float* __restrict__ hp = hbuf + (size_t)rn * ODIM + h * HDIM;
  const float* __restrict__ as = att_src + ((size_t)r * HEADS + h) * HDIM;
  const float* __restrict__ ad = att_dst + ((size_t)r * HEADS + h) * HDIM;
  float ss = 0.f, dd = 0.f;
  #pragma unroll
  for (int c = 0; c < HDIM; c++) {
    float v = hp[c];
    ss += v * as[c];
    dd += v * ad[c];
  }
  a_s[idx] = ss;
  a_d[idx] = dd;
}

// ---------------------------------------------------------------------------
// 3) init: zero out; seed smax with self-loop score; zero denom
// ---------------------------------------------------------------------------
__global__ void rgat_init(float* __restrict__ out, unsigned* __restrict__ smax,
                          float* __restrict__ denom,
                          const float* __restrict__ a_s,
                          const float* __restrict__ a_d) {
  const int idx = blockIdx.x * blockDim.x + threadIdx.x;
  if (idx < NNODES * ODIM) out[idx] = 0.f;
  if (idx < NREL * NNODES * HEADS) {
    smax[idx]  = ford(leaky(a_s[idx] + a_d[idx]));  // self loop: src==dst
    denom[idx] = 0.f;
  }
}

// ---------------------------------------------------------------------------
// 4) per-edge running max (only the edge's own relation; masked relations
//    contribute exactly zero in the reference, so skipping them is exact)
// ---------------------------------------------------------------------------
__global__ void rgat_edge_max(const int* __restrict__ edge_index,
                              const int* __restrict__ edge_type,
                              const float* __restrict__ a_s,
                              const float* __restrict__ a_d,
                              unsigned* __restrict__ smax) {
  const int idx = blockIdx.x * blockDim.x + threadIdx.x;
  if (idx >= NEDGES * HEADS) return;
  const int e = idx >> 2, h = idx & (HEADS - 1);
  const int s = edge_index[e];
  const int d = edge_index[NEDGES + e];
  const int r = edge_type[e];
  const float sc =
      leaky(a_s[((size_t)r * NNODES + s) * HEADS + h] +
            a_d[((size_t)r * NNODES + d) * HEADS + h]);
  atomicMax(&smax[((size_t)r * NNODES + d) * HEADS + h], ford(sc));
}

// ---------------------------------------------------------------------------
// 5) per-edge denom accumulation
// ---------------------------------------------------------------------------
__global__ void rgat_edge_denom(const int* __restrict__ edge_index,
                                const int* __restrict__ edge_type,
                                const float* __restrict__ a_s,
                                const float* __restrict__ a_d,
                                const unsigned* __restrict__ smax,
                                float* __restrict__ denom) {
  const int idx = blockIdx.x * blockDim.x + threadIdx.x;
  if (idx >= NEDGES * HEADS) return;
  const int e = idx >> 2, h = idx & (HEADS - 1);
  const int s = edge_index[e];
  const int d = edge_index[NEDGES + e];
  const int r = edge_type[e];
  const size_t slot = ((size_t)r * NNODES + d) * HEADS + h;
  const float sc =
      leaky(a_s[((size_t)r * NNODES + s) * HEADS + h] + a_d[slot]);
  atomicAdd(&denom[slot], __expf(sc - funord(smax[slot])));
}

// ---------------------------------------------------------------------------
// 6) self-loop denom (one writer per slot -> plain RMW)
// ---------------------------------------------------------------------------
__global__ void rgat_self_denom(const float* __restrict__ a_s,
                                const float* __restrict__ a_d,
                                const unsigned* __restrict__ smax,
                                float* __restrict__ denom) {
  const int idx = blockIdx.x * blockDim.x + threadIdx.x;
  if (idx >= NREL * NNODES * HEADS) return;
  const float sc = leaky(a_s[idx] + a_d[idx]);
  denom[idx] += __expf(sc - funord(smax[idx]));
}

// ---------------------------------------------------------------------------
// 7) edge aggregation: 128 threads per edge (2 edges / block).
//    Coalesced 512B gather of h[src], coalesced f32 atomic scatter to out.
// ---------------------------------------------------------------------------
__global__ __launch_bounds__(256) void rgat_edge_agg(
    const int* __restrict__ edge_index, const int* __restrict__ edge_type,
    const float* __restrict__ a_s, const float* __restrict__ a_d,
    const unsigned* __restrict__ smax, const float* __restrict__ denom,
    const float* __restrict__ hbuf, const float* __restrict__ rel_w,
    float* __restrict__ out) {
  const int t   = threadIdx.x;
  const long e  = (long)blockIdx.x * 2 + (t >> 7);
  const int col = t & (ODIM - 1);
  const int h   = col >> 5;
  if (e >= NEDGES) return;
  const int s = edge_index[e];
  const int d = edge_index[NEDGES + e];
  const int r = edge_type[e];

  const size_t slot = ((size_t)r * NNODES + d) * HEADS + h;
  const float sc =
      leaky(a_s[((size_t)r * NNODES + s) * HEADS + h] + a_d[slot]);
  const float alpha = __expf(sc - funord(smax[slot])) / denom[slot];

  float w[NREL];
  relw(rel_w, w);

  const float msg = w[r] * alpha * hbuf[((size_t)r * NNODES + s) * ODIM + col];
  atomicAdd(&out[(size_t)d * ODIM + col], msg);
}

// ---------------------------------------------------------------------------
// 8) self loops + bias epilogue (single writer per element -> plain RMW)
// ---------------------------------------------------------------------------
__global__ void rgat_self_agg(float* __restrict__ out,
                              const float* __restrict__ hbuf,
                              const float* __restrict__ a_s,
                              const float* __restrict__ a_d,
                              const unsigned* __restrict__ smax,
                              const float* __restrict__ denom,
                              const float* __restrict__ bias,
                              const float* __restrict__ rel_w) {
  const int idx = blockIdx.x * blockDim.x + threadIdx.x;
  if (idx >= NNODES * ODIM) return;
  const int col = idx & (ODIM - 1);
  const int n   = idx >> 7;
  const int h   = col >> 5;

  float w[NREL];
  relw(rel_w, w);

  float acc = out[idx];
  #pragma unroll
  for (int r = 0; r < NREL; r++) {
    const size_t slot = ((size_t)r * NNODES + n) * HEADS + h;
    const float sc    = leaky(a_s[slot] + a_d[slot]);
    const float alpha = __expf(sc - funord(smax[slot])) / denom[slot];
    acc += w[r] * (alpha * hbuf[((size_t)r * NNODES + n) * ODIM + col] +
                   bias[r * ODIM + col]);
  }
  out[idx] = acc;
}

// ---------------------------------------------------------------------------
extern "C" void kernel_launch(void* const* d_in, const int* in_sizes, int n_in,
                              void* d_out, int out_size, void* d_ws,
                              size_t ws_size, hipStream_t stream) {
  const float* x        = (const float*)d_in[0];
  const float* W        = (const float*)d_in[1];
  const float* att_src  = (const float*)d_in[2];
  const float* att_dst  = (const float*)d_in[3];
  const float* bias     = (const float*)d_in[4];
  const float* rel_w    = (const float*)d_in[5];
  const int*   edge_idx = (const int*)d_in[6];
  const int*   edge_ty  = (const int*)d_in[7];
  float*       out      = (float*)d_out;

  // workspace carve-up (floats): h (19.2M) | a_s (600K) | a_d | smax | denom
  float*    hbuf  = (float*)d_ws;
  float*    a_s   = hbuf + (size_t)NREL * NNODES * ODIM;
  float*    a_d   = a_s + (size_t)NREL * NNODES * HEADS;
  unsigned* smax  = (unsigned*)(a_d + (size_t)NREL * NNODES * HEADS);
  float*    denom = (float*)(smax + (size_t)NREL * NNODES * HEADS);

  const int nrh  = NREL * NNODES * HEADS;   // 600,000
  const int nout = NNODES * ODIM;           // 6,400,000
  const int neh  = NEDGES * HEADS;          // 6,400,000

  rgat_gemm<<<dim3(NNODES / 16, NREL), 256, 0, stream>>>(x, W, hbuf);
  rgat_att<<<(nrh + 255) / 256, 256, 0, stream>>>(hbuf, att_src, att_dst, a_s, a_d);
  rgat_init<<<(nout + 255) / 256, 256, 0, stream>>>(out, smax, denom, a_s, a_d);
  rgat_edge_max<<<(neh + 255) / 256, 256, 0, stream>>>(edge_idx, edge_ty, a_s, a_d, smax);
  rgat_edge_denom<<<(neh + 255) / 256, 256, 0, stream>>>(edge_idx, edge_ty, a_s, a_d, smax, denom);
  rgat_self_denom<<<(nrh + 255) / 256, 256, 0, stream>>>(a_s, a_d, smax, denom);
  rgat_edge_agg<<<NEDGES / 2, 256, 0, stream>>>(edge_idx, edge_ty, a_s, a_d,
                                                smax, denom, hbuf, rel_w, out);
  rgat_self_agg<<<(nout + 255) / 256, 256, 0, stream>>>(out, hbuf, a_s, a_d,
                                                        smax, denom, bias, rel_w);
}